// FilterDetection_52235392254191
// MI455X (gfx1250) — compile-verified
//
#include <hip/hip_runtime.h>

#define B_      8
#define N_      50000
#define CDIM    81
#define NCLS    80
#define KSEL    500
#define KPAD    512
#define PROP    100
#define TH2     0.05f
#define IOU_TH  0.5f
#define MAXRATIO 4.135166556742356f

typedef __attribute__((ext_vector_type(2))) float v2f;
typedef __attribute__((ext_vector_type(8))) float v8f;

__device__ __forceinline__ unsigned int mono_key(float s) {
    unsigned int u = __float_as_uint(s);
    return (u & 0x80000000u) ? ~u : (u | 0x80000000u);
}

// ---------------------------------------------------------------------------
// Kernel 1: delta2bbox decode. One thread per (b, n).
// ---------------------------------------------------------------------------
__global__ void decode_kernel(const float* __restrict__ deltas,
                              const float* __restrict__ props,
                              float* __restrict__ boxes) {
    size_t i = (size_t)blockIdx.x * blockDim.x + threadIdx.x;
    if (i >= (size_t)B_ * N_) return;
    const float4 d = ((const float4*)deltas)[i];
    const float4 p = ((const float4*)props)[i];
    float dx = d.x * 0.1f, dy = d.y * 0.1f;
    float dw = fminf(fmaxf(d.z * 0.2f, -MAXRATIO), MAXRATIO);
    float dh = fminf(fmaxf(d.w * 0.2f, -MAXRATIO), MAXRATIO);
    float pw = p.z - p.x, ph = p.w - p.y;
    float px = p.x + 0.5f * pw, py = p.y + 0.5f * ph;
    float cx = px + pw * dx, cy = py + ph * dy;
    float w = pw * __expf(dw), h = ph * __expf(dh);
    float4 o;
    o.x = fminf(fmaxf(cx - 0.5f * w, 0.f), 1.f);
    o.y = fminf(fmaxf(cy - 0.5f * h, 0.f), 1.f);
    o.z = fminf(fmaxf(cx + 0.5f * w, 0.f), 1.f);
    o.w = fminf(fmaxf(cy + 0.5f * h, 0.f), 1.f);
    ((float4*)boxes)[i] = o;
}

// ---------------------------------------------------------------------------
// Kernel 2: per-(b,c) top-500 selection.
// Streams the strided score column into LDS with CDNA5 async-to-LDS loads,
// then 4-level radix select + 512-wide bitonic sort, all in LDS.
// ---------------------------------------------------------------------------
__global__ __launch_bounds__(1024) void select_topk_kernel(
        const float* __restrict__ y,
        float* __restrict__ top_s, int* __restrict__ top_i) {
    __shared__ float ssc[N_];                 // 200 KB score column
    __shared__ unsigned int hist[256];
    __shared__ unsigned long long skey[KPAD];
    __shared__ int eqbuf[64];
    __shared__ unsigned int s_cnt, s_eqcnt, s_pref, s_need;
    __shared__ int s_takeall;

    const int tid = threadIdx.x;
    const int b = blockIdx.x / NCLS;
    const int c = (blockIdx.x % NCLS) + 1;    // skip IGNORE_LABEL=0

    // --- async gather of column y[b, :, c] into LDS (CDNA5 ASYNC path) ---
    const float* colbase = y + (size_t)b * N_ * CDIM + c;
    for (int n = tid; n < N_; n += 1024) {
        unsigned int lds_off = (unsigned int)(size_t)(&ssc[n]);
        const float* g = colbase + (size_t)n * CDIM;
        asm volatile("global_load_async_to_lds_b32 %0, %1, off"
                     :: "v"(lds_off), "v"(g) : "memory");
    }
    asm volatile("s_wait_asynccnt 0" ::: "memory");
    __syncthreads();

    if (tid == 0) { s_takeall = 0; s_need = KSEL; s_pref = 0; }
    __syncthreads();

    // --- 4-level (byte-wise) radix select for the 500 largest keys ---
    for (int lvl = 0; lvl < 4; ++lvl) {
        int shift = 24 - 8 * lvl;
        for (int i = tid; i < 256; i += 1024) hist[i] = 0;
        __syncthreads();
        if (!s_takeall) {
            unsigned int prefv = s_pref;
            for (int n = tid; n < N_; n += 1024) {
                float s = ssc[n];
                if (s > TH2) {
                    unsigned int key = __float_as_uint(s);
                    bool match = (lvl == 0) || ((key >> (shift + 8)) == prefv);
                    if (match) atomicAdd(&hist[(key >> shift) & 255u], 1u);
                }
            }
        }
        __syncthreads();
        if (tid == 0 && !s_takeall) {
            unsigned int need = s_need, acc = 0;
            int t = -1;
            for (int v = 255; v >= 0; --v) {
                if (acc + hist[v] >= need) { t = v; break; }
                acc += hist[v];
            }
            if (t < 0) {
                s_takeall = 1;                 // fewer than `need` pass: take all
            } else {
                s_pref = (s_pref << 8) | (unsigned int)t;
                s_need = need - acc;           // remaining to take from equal bin
            }
        }
        __syncthreads();
    }

    if (tid == 0) { s_cnt = 0; s_eqcnt = 0; }
    __syncthreads();
    const unsigned int T = s_pref;
    const int takeall = s_takeall;
    const unsigned int needEq = s_need;

    // --- collection pass ---
    for (int n = tid; n < N_; n += 1024) {
        float s = ssc[n];
        if (s > TH2) {
            unsigned int key = __float_as_uint(s);
            if (takeall || key > T) {
                unsigned int p = atomicAdd(&s_cnt, 1u);
                if (p < KPAD)
                    skey[p] = ((unsigned long long)(key | 0x80000000u) << 32)
                              | (0xFFFFFFFFu - (unsigned int)n);
            } else if (key == T) {
                unsigned int e = atomicAdd(&s_eqcnt, 1u);
                if (e < 64) eqbuf[e] = n;
            }
        }
    }
    __syncthreads();

    // append ties at the exact threshold key, lowest original index first
    if (tid == 0 && !takeall) {
        unsigned int e = s_eqcnt < 64u ? s_eqcnt : 64u;
        for (unsigned int r = 0; r < needEq && r < e; ++r) {
            int best = -1, bidx = 0x7FFFFFFF;
            for (unsigned int q = 0; q < e; ++q)
                if (eqbuf[q] >= 0 && eqbuf[q] < bidx) { bidx = eqbuf[q]; best = (int)q; }
            if (best < 0) break;
            eqbuf[best] = -1;
            unsigned int p = atomicAdd(&s_cnt, 1u);
            if (p < KPAD)
                skey[p] = ((unsigned long long)(T | 0x80000000u) << 32)
                          | (0xFFFFFFFFu - (unsigned int)bidx);
        }
    }
    __syncthreads();
    unsigned int cnt = s_cnt < (unsigned)KPAD ? s_cnt : (unsigned)KPAD;
    for (int p = tid; p < KPAD; p += 1024)
        if ((unsigned)p >= cnt) skey[p] = 0ull;    // pads (score -1) sort last
    __syncthreads();

    // --- bitonic sort, descending by (score, then lowest index) ---
    for (int k = 2; k <= KPAD; k <<= 1)
        for (int j = k >> 1; j > 0; j >>= 1) {
            __syncthreads();
            for (int t = tid; t < KPAD; t += 1024) {
                int ixj = t ^ j;
                if (ixj > t) {
                    unsigned long long a = skey[t], bb = skey[ixj];
                    bool desc = ((t & k) == 0);
                    if (desc ? (a < bb) : (a > bb)) { skey[t] = bb; skey[ixj] = a; }
                }
            }
        }
    __syncthreads();

    size_t base = (size_t)blockIdx.x * KPAD;
    for (int r = tid; r < KPAD; r += 1024) {
        unsigned long long kk = skey[r];
        float sc; int idx;
        if (kk >> 63) {
            sc = __uint_as_float((unsigned int)(kk >> 32) ^ 0x80000000u);
            idx = (int)(0xFFFFFFFFu - (unsigned int)kk);
        } else { sc = -1.0f; idx = 0; }
        top_s[base + r] = sc;
        top_i[base + r] = idx;
    }
}

// ---------------------------------------------------------------------------
// Kernel 3: per-(b,c) IoU matrix + greedy NMS.
// union = (area_i + area_j) - inter; the (area_i + area_j) pairwise matrix is
// computed with V_WMMA_F32_16X16X4_F32 as a rank-2 outer product:
//   A[m][:] = [area_i(m), 1, 0, 0],  B[:][n] = [1, area_j(n), 0, 0]
// IoU threshold applied divide-free: inter > IOU_TH * max(union, 1e-9).
// Column box bj is k-invariant: loaded once per tile; bi loaded
// unconditionally (indices always < KPAD) so the loop body is branch-free
// except for the ds_or candidate write.
// ---------------------------------------------------------------------------
__global__ __launch_bounds__(512) void nms_kernel(
        const float* __restrict__ boxes,
        float* __restrict__ top_s, const int* __restrict__ top_i) {
    __shared__ float4 sbox[KPAD];
    __shared__ float sarea[KPAD];
    __shared__ float sscr[KPAD];
    __shared__ unsigned int smask[KPAD * 16];   // suppression candidates per row
    __shared__ unsigned int ssupp[16];

    const int tid = threadIdx.x;
    const int b = blockIdx.x / NCLS;
    const size_t base = (size_t)blockIdx.x * KPAD;

    for (int r = tid; r < KPAD; r += 512) {
        float sc = top_s[base + r];
        int idx = top_i[base + r];
        float4 bx = make_float4(0.f, 0.f, 0.f, 0.f);
        if (r < KSEL && sc > TH2)
            bx = *(const float4*)(boxes + ((size_t)b * N_ + idx) * 4);
        sbox[r] = bx;
        sarea[r] = (bx.z - bx.x) * (bx.w - bx.y);
        sscr[r] = sc;
    }
    for (int m = tid; m < KPAD * 16; m += 512) smask[m] = 0u;
    if (tid < 16) ssupp[tid] = 0u;
    __syncthreads();

    const int wave = tid >> 5, lane = tid & 31;
    const int lo = lane & 15;
    const int hi = (lane >> 4) << 3;
    for (int tile = wave; tile < 32 * 32; tile += 16) {
        int ti = tile >> 5, tj = tile & 31;
        float ai = sarea[ti * 16 + lo];
        float aj = sarea[tj * 16 + lo];
        bool low = (lane < 16);
        v2f a, bm;
        a.x = low ? ai : 0.0f;   a.y = low ? 1.0f : 0.0f;     // A: K0 | K1 (lanes<16), K2|K3 = 0
        bm.x = low ? 1.0f : 0.0f; bm.y = low ? aj : 0.0f;     // B: K0 row = 1, K1 row = area_j
        v8f acc = {0.f,0.f,0.f,0.f,0.f,0.f,0.f,0.f};
        acc = __builtin_amdgcn_wmma_f32_16x16x4_f32(
                false, a, false, bm, (short)0, acc, false, false);

        const int j = tj * 16 + lo;           // k-invariant column index
        const float4 bj = sbox[j];            // load once per tile
        const unsigned int jbit = 1u << (j & 31);
        const int jword = j >> 5;
#pragma unroll
        for (int k = 0; k < 8; ++k) {
            int i = ti * 16 + hi + k;
            float4 bi = sbox[i];              // always in-range (i < 512)
            float iw = fminf(bi.z, bj.z) - fmaxf(bi.x, bj.x);
            float ih = fminf(bi.w, bj.w) - fmaxf(bi.y, bj.y);
            float inter = fmaxf(iw, 0.f) * fmaxf(ih, 0.f);
            float uni = acc[k] - inter;       // area_i + area_j - inter (WMMA)
            // iou > 0.5  <=>  inter > 0.5 * max(uni, 1e-9)   (divide-free)
            bool hit = (inter > IOU_TH * fmaxf(uni, 1e-9f))
                       && (i < KSEL) && (j < KSEL) && (i != j);
            if (hit) atomicOr(&smask[i * 16 + jword], jbit);
        }
    }
    __syncthreads();

    // greedy propagation (matches reference fori_loop): one wave, 16 word-lanes
    if (tid < 16) {
        const int l = tid;
        for (int i = 0; i < KSEL; ++i) {
            unsigned int wi = ssupp[i >> 5];
            bool sup = (wi >> (i & 31)) & 1u;
            if (!sup) {
                unsigned int row = smask[i * 16 + l];
                int wrd = i >> 5, bit = i & 31;
                unsigned int gt;
                if (l < wrd)      gt = 0u;
                else if (l > wrd) gt = ~0u;
                else              gt = (bit == 31) ? 0u : (~0u << (bit + 1));
                ssupp[l] |= row & gt;
            }
        }
    }
    __syncthreads();

    for (int r = tid; r < KPAD; r += 512) {
        bool sup = (ssupp[r >> 5] >> (r & 31)) & 1u;
        float sc = sscr[r];
        top_s[base + r] = (r < KSEL && !sup && sc > TH2) ? sc : -1.0f;
    }
}

// ---------------------------------------------------------------------------
// Kernel 4: per-image top-100 over the 80*512 kept scores + output gather.
// ---------------------------------------------------------------------------
__global__ __launch_bounds__(1024) void final_topk_kernel(
        const float* __restrict__ y, const float* __restrict__ boxes,
        const float* __restrict__ top_s, const int* __restrict__ top_i,
        float* __restrict__ out) {
    __shared__ float ks[NCLS * KPAD];           // 160 KB
    __shared__ unsigned long long red[1024];
    __shared__ int winner[PROP];
    __shared__ float winsc[PROP];

    const int tid = threadIdx.x;
    const int b = blockIdx.x;
    const size_t ibase = (size_t)b * NCLS * KPAD;

    for (int f = tid; f < NCLS * KPAD; f += 1024) ks[f] = top_s[ibase + f];
    __syncthreads();

    for (int it = 0; it < PROP; ++it) {
        unsigned long long best = 0ull;
        for (int f = tid; f < NCLS * KPAD; f += 1024) {
            unsigned long long kk =
                ((unsigned long long)mono_key(ks[f]) << 32)
                | (0xFFFFFFFFu - (unsigned int)f);
            if (kk > best) best = kk;
        }
        red[tid] = best;
        __syncthreads();
        for (int st = 512; st > 0; st >>= 1) {
            if (tid < st && red[tid + st] > red[tid]) red[tid] = red[tid + st];
            __syncthreads();
        }
        if (tid == 0) {
            int f0 = (int)(0xFFFFFFFFu - (unsigned int)red[0]);
            winner[it] = f0;
            winsc[it] = ks[f0];
            ks[f0] = -2.0f;                    // exclude from next iterations
        }
        __syncthreads();
    }

    for (int e = tid; e < PROP * (CDIM + 4); e += 1024) {
        int it = e / (CDIM + 4), col = e % (CDIM + 4);
        int f0 = winner[it];
        float valid = (winsc[it] > TH2) ? 1.0f : 0.0f;
        int orig = top_i[ibase + f0];
        float v;
        if (col < CDIM) v = y[((size_t)b * N_ + orig) * CDIM + col] * valid;
        else            v = boxes[((size_t)b * N_ + orig) * 4 + (col - CDIM)] * valid;
        out[(size_t)b * PROP * (CDIM + 4) + e] = v;
    }
}

// ---------------------------------------------------------------------------
extern "C" void kernel_launch(void* const* d_in, const int* in_sizes, int n_in,
                              void* d_out, int out_size, void* d_ws, size_t ws_size,
                              hipStream_t stream) {
    const float* y_pred    = (const float*)d_in[0];   // [8,50000,81]
    const float* bbox_pred = (const float*)d_in[1];   // [8,50000,4]
    const float* proposals = (const float*)d_in[2];   // [8,50000,4]
    float* out = (float*)d_out;                       // [8,100,85]

    char* ws = (char*)d_ws;
    float* boxes = (float*)ws;                                   // B*N*4 f32
    size_t off = (size_t)B_ * N_ * 4 * sizeof(float);
    float* top_s = (float*)(ws + off);                           // B*80*512 f32
    off += (size_t)B_ * NCLS * KPAD * sizeof(float);
    int* top_i = (int*)(ws + off);                               // B*80*512 i32
    (void)ws_size; (void)in_sizes; (void)n_in; (void)out_size;

    const int total = B_ * N_;
    decode_kernel<<<(total + 255) / 256, 256, 0, stream>>>(bbox_pred, proposals, boxes);
    select_topk_kernel<<<B_ * NCLS, 1024, 0, stream>>>(y_pred, top_s, top_i);
    nms_kernel<<<B_ * NCLS, 512, 0, stream>>>(boxes, top_s, top_i);
    final_topk_kernel<<<B_, 1024, 0, stream>>>(y_pred, boxes, top_s, top_i, out);
}